// Grid_10479720202790
// MI455X (gfx1250) — compile-verified
//
#include <hip/hip_runtime.h>
#include <cstdint>

// Hash-grid trilinear interpolation (Instant-NGP single level), gfx1250.
// T = 2^19 entries x 2 features (4 MB -> L2 resident), N = 2,097,152 points.

#define HASH_T    524288u
#define HASH_MASK (HASH_T - 1u)
#define PRIME1    2654435761u
#define PRIME2    805459861u
// RES - 1 = 16 * 32^(8/15) - 1 = 100.59366732596477
#define SCALE     100.59366732596477f

#define PTS_PER_BLOCK 512
#define PTS_PER_WAVE  64

__device__ __forceinline__ float2 eval_point(float x0, float x1, float x2,
                                             const float2* __restrict__ tab) {
    float xs0 = x0 * SCALE, xs1 = x1 * SCALE, xs2 = x2 * SCALE;
    float fl0 = floorf(xs0), fl1 = floorf(xs1), fl2 = floorf(xs2);
    float w0 = xs0 - fl0, w1 = xs1 - fl1, w2 = xs2 - fl2;
    // coords are non-negative; mod 2^19 of the int64 hash == 32-bit mul + mask
    uint32_t hf0 = (uint32_t)(int)fl0;
    uint32_t hf1 = (uint32_t)(int)fl1 * PRIME1;
    uint32_t hf2 = (uint32_t)(int)fl2 * PRIME2;
    uint32_t hc0 = (uint32_t)(int)ceilf(xs0);
    uint32_t hc1 = (uint32_t)(int)ceilf(xs1) * PRIME1;
    uint32_t hc2 = (uint32_t)(int)ceilf(xs2) * PRIME2;
    // corner k: dim0<-bit2, dim1<-bit1, dim2<-bit0 (matches _CORNER_MASK)
    float2 v0 = tab[(hf0 ^ hf1 ^ hf2) & HASH_MASK];
    float2 v1 = tab[(hf0 ^ hf1 ^ hc2) & HASH_MASK];
    float2 v2 = tab[(hf0 ^ hc1 ^ hf2) & HASH_MASK];
    float2 v3 = tab[(hf0 ^ hc1 ^ hc2) & HASH_MASK];
    float2 v4 = tab[(hc0 ^ hf1 ^ hf2) & HASH_MASK];
    float2 v5 = tab[(hc0 ^ hf1 ^ hc2) & HASH_MASK];
    float2 v6 = tab[(hc0 ^ hc1 ^ hf2) & HASH_MASK];
    float2 v7 = tab[(hc0 ^ hc1 ^ hc2) & HASH_MASK];
    float om0 = 1.0f - w0, om1 = 1.0f - w1, om2 = 1.0f - w2;
    float p00x = v0.x * om0 + v4.x * w0, p00y = v0.y * om0 + v4.y * w0;
    float p01x = v1.x * om0 + v5.x * w0, p01y = v1.y * om0 + v5.y * w0;
    float p10x = v2.x * om0 + v6.x * w0, p10y = v2.y * om0 + v6.y * w0;
    float p11x = v3.x * om0 + v7.x * w0, p11y = v3.y * om0 + v7.y * w0;
    float p0x = p00x * om1 + p10x * w1, p0y = p00y * om1 + p10y * w1;
    float p1x = p01x * om1 + p11x * w1, p1y = p01y * om1 + p11y * w1;
    float2 r;
    r.x = p0x * om2 + p1x * w2;
    r.y = p0y * om2 + p1y * w2;
    return r;
}

// Main kernel: full blocks only (npoints_full = blocks*512).
// Each wave32 async-copies its 64 points of X (768 B) into its own LDS slice
// (GLOBAL_LOAD_ASYNC_TO_LDS_B64, ASYNCcnt-tracked), waits s_wait_asynccnt 0,
// then each lane evaluates 2 points and stores one 16-B float4.
__global__ __launch_bounds__(256) void hashgrid_main(const float* __restrict__ X,
                                                     const float2* __restrict__ tab,
                                                     float4* __restrict__ out4) {
    __shared__ float sx[PTS_PER_BLOCK * 3];  // 6144 bytes
    const int lane = threadIdx.x & 31;
    const int wave = threadIdx.x >> 5;
    const int pw = blockIdx.x * PTS_PER_BLOCK + wave * PTS_PER_WAVE;  // wave's first point

    // --- async stage: 768 bytes = 3 rounds x 32 lanes x 8 bytes ---
    {
        // low 32 bits of a flat shared pointer == DS-usable LDS byte address
        uint32_t lds_base = (uint32_t)(size_t)(&sx[wave * (PTS_PER_WAVE * 3)]);
        const char* gbase = (const char*)X + (size_t)pw * 12u;
#pragma unroll
        for (int i = 0; i < 3; ++i) {
            uint32_t lds_addr = lds_base + (uint32_t)(i * 256 + lane * 8);
            uint64_t gaddr = (uint64_t)(size_t)(gbase + i * 256 + lane * 8);
            asm volatile("global_load_async_to_lds_b64 %0, %1, off"
                         :: "v"(lds_addr), "v"(gaddr)
                         : "memory");
        }
        asm volatile("s_wait_asynccnt 0" ::: "memory");
        // each wave reads only its own staged slice -> no workgroup barrier
    }

    const float* my = &sx[wave * (PTS_PER_WAVE * 3) + lane * 6];
    float a0 = my[0], a1 = my[1], a2 = my[2];
    float b0 = my[3], b1 = my[4], b2 = my[5];

    float2 ra = eval_point(a0, a1, a2, tab);
    float2 rb = eval_point(b0, b1, b2, tab);

    // points q0 = pw + 2*lane, q0+1 -> out floats [2*q0 .. 2*q0+3], 16-B aligned
    out4[(pw >> 1) + lane] = make_float4(ra.x, ra.y, rb.x, rb.y);
}

// Tail kernel (generic remainder path; not launched for N = 2,097,152).
__global__ void hashgrid_tail(const float* __restrict__ X,
                              const float2* __restrict__ tab,
                              float2* __restrict__ out,
                              int start, int npoints) {
    int p = start + blockIdx.x * blockDim.x + threadIdx.x;
    if (p >= npoints) return;
    float2 r = eval_point(X[3 * p + 0], X[3 * p + 1], X[3 * p + 2], tab);
    out[p] = r;
}

extern "C" void kernel_launch(void* const* d_in, const int* in_sizes, int n_in,
                              void* d_out, int out_size, void* d_ws, size_t ws_size,
                              hipStream_t stream) {
    const float* X = (const float*)d_in[0];          // (N, 3) float32
    const float2* tab = (const float2*)d_in[1];      // (T, 2) float32
    const int npoints = in_sizes[0] / 3;             // 2,097,152

    const int blocks = npoints / PTS_PER_BLOCK;      // 4096 full blocks
    if (blocks > 0) {
        hashgrid_main<<<blocks, 256, 0, stream>>>(X, tab, (float4*)d_out);
    }
    const int done = blocks * PTS_PER_BLOCK;
    const int rem = npoints - done;
    if (rem > 0) {
        int tblocks = (rem + 255) / 256;
        hashgrid_tail<<<tblocks, 256, 0, stream>>>(X, tab, (float2*)d_out, done, npoints);
    }
}